// NoCrossSAGEEncoderModel_44504451121589
// MI455X (gfx1250) — compile-verified
//
#include <hip/hip_runtime.h>
#include <math.h>

typedef float v2f __attribute__((ext_vector_type(2)));
typedef float v8f __attribute__((ext_vector_type(8)));

#define D_FEAT 128
#define H_FEAT 16

// ---------------- utility: zero a float buffer ----------------
__global__ void zero_kernel(float* __restrict__ p, long long n) {
    long long i = (long long)blockIdx.x * blockDim.x + threadIdx.x;
    if (i < n) p[i] = 0.0f;
}

// ---------------- degree ----------------
__global__ void degree_kernel(const int* __restrict__ dst, float* __restrict__ deg, int E) {
    int e = blockIdx.x * blockDim.x + threadIdx.x;
    if (e < E) atomicAdd(&deg[dst[e]], 1.0f);
}

__global__ void invdeg_kernel(float* __restrict__ deg, int n) {
    int i = blockIdx.x * blockDim.x + threadIdx.x;
    if (i < n) deg[i] = 1.0f / fmaxf(deg[i], 1.0f);
}

// ---------------- GEMM1: Y = X @ Wl, Z = X @ Wr  (X:[n,128], W:[128,16]) ----------------
// One wave per 16x16 output tile; K=128 via 32 x V_WMMA_F32_16X16X4_F32.
__global__ void gemm1_wmma(const float* __restrict__ X,
                           const float* __restrict__ Wl,
                           const float* __restrict__ Wr,
                           float* __restrict__ Y,
                           float* __restrict__ Z,
                           int n) {
    const int lane = threadIdx.x & 31;
    const int wave = blockIdx.x * (blockDim.x >> 5) + (threadIdx.x >> 5);
    const int ntm  = (n + 15) >> 4;          // row tiles
    const int mt    = wave >> 1;             // which 16-row tile
    const int which = wave & 1;              // 0 -> Wl/Y, 1 -> Wr/Z
    if (mt >= ntm) return;                   // wave-uniform: EXEC stays all-1 for WMMA

    const float* __restrict__ W = which ? Wr : Wl;
    float* __restrict__ O       = which ? Z  : Y;

    const int m  = lane & 15;                // A: row within tile ; B/C: column
    const int kh = lane >> 4;                // K half-select (ISA 16x4 f32 A layout)
    const size_t row = (size_t)(mt * 16 + m);
    const float* __restrict__ arow = X + row * D_FEAT;

    v8f acc = {};
    #pragma unroll
    for (int k0 = 0; k0 < D_FEAT; k0 += 4) {
        const int kb = k0 + 2 * kh;
        v2f a, b;
        a.x = arow[kb + 0];
        a.y = arow[kb + 1];
        b.x = W[(size_t)(kb + 0) * H_FEAT + m];
        b.y = W[(size_t)(kb + 1) * H_FEAT + m];
        acc = __builtin_amdgcn_wmma_f32_16x16x4_f32(
            /*neg_a=*/false, a, /*neg_b=*/false, b,
            /*c_mod=*/(short)0, acc, /*reuse_a=*/false, /*reuse_b=*/false);
    }
    // C/D layout: VGPR r -> M = r + 8*kh, N = lane&15
    #pragma unroll
    for (int r = 0; r < 8; ++r) {
        const size_t cm = (size_t)(mt * 16 + r + 8 * kh);
        O[cm * H_FEAT + m] = acc[r];
    }
}

// ---------------- scatter-add 16 features per edge ----------------
__global__ void scatter_add16(const float* __restrict__ V,     // [n,16]
                              const int* __restrict__ src,
                              const int* __restrict__ dst,
                              float* __restrict__ OUT,         // [n,16]
                              long long e16) {
    long long idx = (long long)blockIdx.x * blockDim.x + threadIdx.x;
    if (idx >= e16) return;
    const int e = (int)(idx >> 4);
    const int f = (int)(idx & 15);
    const int s = src[e];
    const int d = dst[e];
    atomicAdd(&OUT[(size_t)d * H_FEAT + f], V[(size_t)s * H_FEAT + f]);
}

// ---------------- h = relu(agg*inv_deg + b1 + z) ----------------
__global__ void relu_combine(const float* __restrict__ agg,
                             const float* __restrict__ z,
                             const float* __restrict__ inv,
                             const float* __restrict__ b1,
                             float* __restrict__ h,
                             long long n16) {
    long long idx = (long long)blockIdx.x * blockDim.x + threadIdx.x;
    if (idx >= n16) return;
    const int i = (int)(idx >> 4);
    const int f = (int)(idx & 15);
    const float v = agg[idx] * inv[i] + b1[f] + z[idx];
    h[idx] = fmaxf(v, 0.0f);
}

// ---------------- GEMM2: X2 = [agg*inv | h] @ [W2l ; W2r] + b2 ----------------
// A2:[n,32] (virtual), W2:[32,128]; one wave per 16x16 tile of X2:[n,128].
__global__ void gemm2_wmma(const float* __restrict__ AGG,   // [n,16]
                           const float* __restrict__ Hh,    // [n,16]
                           const float* __restrict__ INV,   // [n]
                           const float* __restrict__ W2l,   // [16,128]
                           const float* __restrict__ W2r,   // [16,128]
                           const float* __restrict__ B2,    // [128]
                           float* __restrict__ X2,          // [n,128]
                           int n) {
    const int lane = threadIdx.x & 31;
    const int wave = blockIdx.x * (blockDim.x >> 5) + (threadIdx.x >> 5);
    const int ntm  = (n + 15) >> 4;
    const int mt = wave >> 3;                // row tile
    const int nt = wave & 7;                 // 8 column tiles of 16 (D=128)
    if (mt >= ntm) return;                   // wave-uniform

    const int m  = lane & 15;
    const int kh = lane >> 4;
    const size_t row = (size_t)(mt * 16 + m);
    const float inv  = INV[row];
    const int n0 = nt * 16;

    v8f acc = {};
    #pragma unroll
    for (int k0 = 0; k0 < 32; k0 += 4) {
        v2f a, b;
        #pragma unroll
        for (int v = 0; v < 2; ++v) {
            const int k = k0 + 2 * kh + v;
            float av, bv;
            if (k < 16) {
                av = AGG[row * H_FEAT + k] * inv;
                bv = W2l[(size_t)k * D_FEAT + n0 + m];
            } else {
                av = Hh[row * H_FEAT + (k - 16)];
                bv = W2r[(size_t)(k - 16) * D_FEAT + n0 + m];
            }
            a[v] = av;
            b[v] = bv;
        }
        acc = __builtin_amdgcn_wmma_f32_16x16x4_f32(
            false, a, false, b, (short)0, acc, false, false);
    }
    const float bias = B2[n0 + m];
    #pragma unroll
    for (int r = 0; r < 8; ++r) {
        const size_t cm = (size_t)(mt * 16 + r + 8 * kh);
        X2[cm * D_FEAT + n0 + m] = acc[r] + bias;
    }
}

// ---------------- classifier + per-block BCE partial sums ----------------
__global__ void classifier_kernel(const float* __restrict__ X2,
                                  const int* __restrict__ a1,
                                  const int* __restrict__ a2,
                                  const int* __restrict__ labels,
                                  const float* __restrict__ wc,   // [256]
                                  const float* __restrict__ bc,   // [1]
                                  float* __restrict__ logits,     // d_out + 1
                                  float* __restrict__ partial,    // [gridDim.x]
                                  int B) {
    const int b = blockIdx.x * blockDim.x + threadIdx.x;
    float term = 0.0f;
    if (b < B) {
        const float* __restrict__ r1 = X2 + (size_t)a1[b] * D_FEAT;
        const float* __restrict__ r2 = X2 + (size_t)a2[b] * D_FEAT;
        float s = bc[0];
        #pragma unroll 8
        for (int k = 0; k < D_FEAT; ++k)
            s += r1[k] * wc[k] + r2[k] * wc[D_FEAT + k];
        logits[b] = s;
        const float y = (float)labels[b];
        term = fmaxf(s, 0.0f) - s * y + log1pf(expf(-fabsf(s)));
    }
    __shared__ float red[256];
    red[threadIdx.x] = term;
    __syncthreads();
    for (int off = blockDim.x >> 1; off > 0; off >>= 1) {
        if (threadIdx.x < (unsigned)off) red[threadIdx.x] += red[threadIdx.x + off];
        __syncthreads();
    }
    if (threadIdx.x == 0) partial[blockIdx.x] = red[0];
}

__global__ void finalize_loss(const float* __restrict__ partial, int nblocks,
                              float* __restrict__ out0, float invB) {
    __shared__ float red[256];
    float s = 0.0f;
    for (int i = threadIdx.x; i < nblocks; i += blockDim.x) s += partial[i];
    red[threadIdx.x] = s;
    __syncthreads();
    for (int off = blockDim.x >> 1; off > 0; off >>= 1) {
        if (threadIdx.x < (unsigned)off) red[threadIdx.x] += red[threadIdx.x + off];
        __syncthreads();
    }
    if (threadIdx.x == 0) out0[0] = red[0] * invB;
}

// ---------------- launch ----------------
extern "C" void kernel_launch(void* const* d_in, const int* in_sizes, int n_in,
                              void* d_out, int out_size, void* d_ws, size_t ws_size,
                              hipStream_t stream) {
    const float* X    = (const float*)d_in[0];   // [N,128]
    const float* w1l  = (const float*)d_in[1];   // [128,16]
    const float* b1   = (const float*)d_in[2];   // [16]
    const float* w1r  = (const float*)d_in[3];   // [128,16]
    const float* w2l  = (const float*)d_in[4];   // [16,128]
    const float* b2   = (const float*)d_in[5];   // [128]
    const float* w2r  = (const float*)d_in[6];   // [16,128]
    const float* wc   = (const float*)d_in[7];   // [256]
    const float* bc   = (const float*)d_in[8];   // [1]
    const int*   ei   = (const int*)d_in[9];     // [2,E]
    const int*   a1   = (const int*)d_in[10];    // [B]
    const int*   a2   = (const int*)d_in[11];    // [B]
    const int*   lab  = (const int*)d_in[12];    // [B]

    const int N = in_sizes[0] / D_FEAT;
    const int E = in_sizes[9] / 2;
    const int B = in_sizes[10];
    const int* src = ei;
    const int* dst = ei + E;

    // workspace layout (floats)
    float* ws   = (float*)d_ws;
    size_t off  = 0;
    float* deg  = ws + off; off += (size_t)N;             // deg -> inv_deg (in place)
    float* y1   = ws + off; off += (size_t)N * H_FEAT;    // X @ w1_l
    float* z1   = ws + off; off += (size_t)N * H_FEAT;    // X @ w1_r
    float* agg  = ws + off; off += (size_t)N * H_FEAT;    // reused for both scatters
    float* hbuf = ws + off; off += (size_t)N * H_FEAT;    // relu output
    float* x2   = ws + off; off += (size_t)N * D_FEAT;    // layer-2 output
    float* part = ws + off; off += 256;                   // loss partials

    float* out_loss   = (float*)d_out;
    float* out_logits = (float*)d_out + 1;

    const int TB = 256;
    const long long n16 = (long long)N * H_FEAT;
    const long long e16 = (long long)E * H_FEAT;

    // 1) degrees
    zero_kernel<<<(int)((N + TB - 1) / TB), TB, 0, stream>>>(deg, N);
    zero_kernel<<<(int)((n16 + TB - 1) / TB), TB, 0, stream>>>(agg, n16);
    degree_kernel<<<(E + TB - 1) / TB, TB, 0, stream>>>(dst, deg, E);
    invdeg_kernel<<<(N + TB - 1) / TB, TB, 0, stream>>>(deg, N);

    // 2) projection GEMM (WMMA f32): y1 = X@w1l, z1 = X@w1r
    {
        const int ntm = (N + 15) / 16;
        const int waves = ntm * 2;
        const int blocks = (waves + (TB / 32) - 1) / (TB / 32);
        gemm1_wmma<<<blocks, TB, 0, stream>>>(X, w1l, w1r, y1, z1, N);
    }

    // 3) neighbor mean (post-projection) + relu
    scatter_add16<<<(int)((e16 + TB - 1) / TB), TB, 0, stream>>>(y1, src, dst, agg, e16);
    relu_combine<<<(int)((n16 + TB - 1) / TB), TB, 0, stream>>>(agg, z1, deg, b1, hbuf, n16);

    // 4) second aggregation
    zero_kernel<<<(int)((n16 + TB - 1) / TB), TB, 0, stream>>>(agg, n16);
    scatter_add16<<<(int)((e16 + TB - 1) / TB), TB, 0, stream>>>(hbuf, src, dst, agg, e16);

    // 5) layer-2 GEMM (WMMA f32): x2 = [agg*inv | h] @ [w2l;w2r] + b2
    {
        const int ntm = (N + 15) / 16;
        const int waves = ntm * 8;
        const int blocks = (waves + (TB / 32) - 1) / (TB / 32);
        gemm2_wmma<<<blocks, TB, 0, stream>>>(agg, hbuf, deg, w2l, w2r, b2, x2, N);
    }

    // 6) classifier + deterministic loss reduction
    const int cblocks = (B + TB - 1) / TB;   // 64 for B=16384
    classifier_kernel<<<cblocks, TB, 0, stream>>>(x2, a1, a2, lab, wc, bc,
                                                  out_logits, part, B);
    finalize_loss<<<1, TB, 0, stream>>>(part, cblocks, out_loss, 1.0f / (float)B);
}